// Sparsemax_79972291051848
// MI455X (gfx1250) — compile-verified
//
#include <hip/hip_runtime.h>
#include <stdint.h>

// ---------------------------------------------------------------------------
// Sparsemax over last axis, rows of length 32  ==  one wave32 per row.
// Memory-bound: 1 GiB traffic -> ~44us floor at 23.3 TB/s. Per-row compute is
// a 15-step bitonic sort (ds_swizzle xor butterflies), a 5-step inclusive scan,
// ballot/popcount for the support size, one broadcast, one clip.
// Input staged to LDS with the Tensor Data Mover (double buffered, 8KB tiles),
// synchronized with s_wait_tensorcnt + workgroup barriers.
// ---------------------------------------------------------------------------

#if defined(__AMDGCN__) && __has_builtin(__builtin_amdgcn_tensor_load_to_lds)
#define SPX_TDM 1
#else
#define SPX_TDM 0
#endif

#define SPX_TPB 256          // 8 waves per block
#define SPX_TROWS 64         // rows per tile (8 waves x 8 rows)
#define SPX_TILE_ELEMS 2048  // 64 rows * 32 f32 = 8KB

// lane-xor shuffle via ds_swizzle group-of-32 mode: new_lane = lane ^ J
#define SPX_SWZ(v, J) \
  __int_as_float(__builtin_amdgcn_ds_swizzle(__float_as_int(v), (0x1f | ((J) << 10))))

// One bitonic compare-exchange step. asc = ((lane & K) != 0) gives a network
// whose final K=32 merge is DESCENDING (lane&32 == 0 for all wave32 lanes).
#define SPX_BSTEP(z, lane, K, J)                                   \
  do {                                                             \
    float _o = SPX_SWZ(z, J);                                      \
    bool _asc = ((lane) & (K)) != 0;                               \
    bool _lo = ((lane) & (J)) == 0;                                \
    (z) = (_lo == _asc) ? fminf((z), _o) : fmaxf((z), _o);         \
  } while (0)

// Full sparsemax for one row held one-element-per-lane. Returns clipped value.
__device__ __forceinline__ float sparsemax_row32(float x, int lane) {
#if defined(__AMDGCN__)
  // ---- bitonic sort descending across 32 lanes (15 ds_swizzle steps) ----
  float z = x;
  SPX_BSTEP(z, lane, 2, 1);
  SPX_BSTEP(z, lane, 4, 2);  SPX_BSTEP(z, lane, 4, 1);
  SPX_BSTEP(z, lane, 8, 4);  SPX_BSTEP(z, lane, 8, 2);  SPX_BSTEP(z, lane, 8, 1);
  SPX_BSTEP(z, lane, 16, 8); SPX_BSTEP(z, lane, 16, 4); SPX_BSTEP(z, lane, 16, 2);
  SPX_BSTEP(z, lane, 16, 1);
  SPX_BSTEP(z, lane, 32, 16); SPX_BSTEP(z, lane, 32, 8); SPX_BSTEP(z, lane, 32, 4);
  SPX_BSTEP(z, lane, 32, 2);  SPX_BSTEP(z, lane, 32, 1);

  // ---- inclusive prefix sum of sorted z (Hillis-Steele, 5 steps) ----
  float cs = z;
#pragma unroll
  for (int off = 1; off < 32; off <<= 1) {
    float t = __shfl_up(cs, off, 32);
    cs += (lane >= off) ? t : 0.0f;
  }

  // support test: z - (cumsum-1)/r > 0  <=>  z*r > cumsum - 1   (r = lane+1 > 0)
  bool cond = z * (float)(lane + 1) > (cs - 1.0f);
  unsigned long long m = __ballot(cond);
  int k = __popcll(m);  // k >= 1 always (top element always passes)

  float csk = __shfl(cs, k - 1, 32);      // cumsum of top-k
  float tau = (csk - 1.0f) / (float)k;
  return fmaxf(x - tau, 0.0f);
#else
  (void)lane;
  return x;  // host-pass stub, never executed
#endif
}

// Issue one TDM DMA: 8KB contiguous tile (2048 f32) global -> LDS.
// D# per CDNA5 ISA ch.8: tile == tensor (1-D, fully in-bounds), data_size=4B.
__device__ __forceinline__ void spx_tdm_load_tile(const void* gptr, void* ldsPtr) {
#if SPX_TDM
  typedef unsigned int uv4 __attribute__((ext_vector_type(4)));
  typedef int iv4 __attribute__((ext_vector_type(4)));
  typedef int iv8 __attribute__((ext_vector_type(8)));

  unsigned long long ga = (unsigned long long)(uintptr_t)gptr;
  unsigned lds = (unsigned)(uintptr_t)ldsPtr;  // flat shared addr low 32 = LDS offset

  // group0: count=1 | lds_addr[63:32] | global_addr[120:64] | type=2 [127:126]
  uv4 g0 = {1u, lds, (unsigned)ga, (unsigned)(ga >> 32) | 0x80000000u};

  // group1: wg_mask=0, data_size=2 (4B), no barrier/iterate/pad,
  //         tensor_dim0=2048, tensor_dim1=1, tile_dim0=2048, tile_dim1=1,
  //         tile_dim2=0, tensor_dim0_stride=2048, tensor_dim1_stride=0
  iv8 g1 = {0x20000,
            (int)(2048u << 16),   // tensor_dim0[15:0] at bits 63:48
            (int)(1u << 16),      // tensor_dim0 hi=0 | tensor_dim1 lo=1
            (int)(2048u << 16),   // tensor_dim1 hi=0 | tile_dim0=2048
            1,                    // tile_dim1=1 | tile_dim2=0
            2048,                 // tensor_dim0_stride lo32
            0, 0};
  iv4 zg = {0, 0, 0, 0};  // groups 2/3 unused (<=2D)

#if __clang_major__ >= 23
  iv8 z8 = {0, 0, 0, 0, 0, 0, 0, 0};
  __builtin_amdgcn_tensor_load_to_lds(g0, g1, zg, zg, z8, 0);
#else
  __builtin_amdgcn_tensor_load_to_lds(g0, g1, zg, zg, 0);
#endif
#else
  (void)gptr;
  (void)ldsPtr;
#endif
}

// Main kernel: grid-stride over 64-row tiles. Wave 0 runs a double-buffered
// TDM pipeline into LDS; all 8 waves compute 8 rows each per tile.
__global__ void __launch_bounds__(SPX_TPB)
sparsemax_tdm_kernel(const float* __restrict__ in, float* __restrict__ out, int ntiles) {
  const int lane = threadIdx.x & 31;
  const int wv = threadIdx.x >> 5;

  if ((int)blockIdx.x >= ntiles) return;  // whole block exits uniformly

#if SPX_TDM
  __shared__ float smem[2][SPX_TILE_ELEMS];
  const bool leader = (wv == 0);

  // prologue: prefetch first tile into buffer 0
  if (leader) spx_tdm_load_tile(in + (size_t)blockIdx.x * SPX_TILE_ELEMS, &smem[0][0]);

  int buf = 0;
  for (int tile = blockIdx.x; tile < ntiles; tile += gridDim.x, buf ^= 1) {
    if (leader) __builtin_amdgcn_s_wait_tensorcnt(0);  // tile data landed in LDS
    __syncthreads();  // publish tile; also: everyone done reading buf^1 last iter

    int nxt = tile + gridDim.x;
    if (leader && nxt < ntiles)
      spx_tdm_load_tile(in + (size_t)nxt * SPX_TILE_ELEMS, &smem[buf ^ 1][0]);

    const size_t gbase = (size_t)tile * SPX_TILE_ELEMS;
#pragma unroll
    for (int rr = 0; rr < 8; ++rr) {
      const int rowInTile = wv * 8 + rr;
      float x = smem[buf][rowInTile * 32 + lane];
      out[gbase + (size_t)rowInTile * 32 + lane] = sparsemax_row32(x, lane);
    }
    __syncthreads();  // all reads of smem[buf] done before it is DMA-overwritten
  }
#else
  // fallback (no TDM builtin): direct coalesced global loads, same tiling
  for (int tile = blockIdx.x; tile < ntiles; tile += gridDim.x) {
    const size_t gbase = (size_t)tile * SPX_TILE_ELEMS;
#pragma unroll
    for (int rr = 0; rr < 8; ++rr) {
      const int rowInTile = wv * 8 + rr;
      float x = in[gbase + (size_t)rowInTile * 32 + lane];
      out[gbase + (size_t)rowInTile * 32 + lane] = sparsemax_row32(x, lane);
    }
  }
#endif
}

// Tail / generic kernel: one wave per row, grid-stride over waves.
__global__ void __launch_bounds__(SPX_TPB)
sparsemax_rowwave_kernel(const float* __restrict__ in, float* __restrict__ out,
                         long long rowStart, long long rowEnd) {
  const int lane = threadIdx.x & 31;
  const long long gw = (long long)((blockIdx.x * blockDim.x + threadIdx.x) >> 5);
  const long long nw = (long long)((gridDim.x * blockDim.x) >> 5);
  for (long long r = rowStart + gw; r < rowEnd; r += nw) {
    float x = in[r * 32 + lane];
    out[r * 32 + lane] = sparsemax_row32(x, lane);
  }
}

extern "C" void kernel_launch(void* const* d_in, const int* in_sizes, int n_in,
                              void* d_out, int out_size, void* d_ws, size_t ws_size,
                              hipStream_t stream) {
  (void)n_in; (void)out_size; (void)d_ws; (void)ws_size;
  const float* x = (const float*)d_in[0];
  float* out = (float*)d_out;

  const long long n = (long long)in_sizes[0];
  const long long rows = n / 32;           // 4,194,304 for the reference shape
  const long long tiles = rows / SPX_TROWS;

  if (tiles > 0) {
    long long grid = tiles < 8192 ? tiles : 8192;
    sparsemax_tdm_kernel<<<dim3((unsigned)grid), dim3(SPX_TPB), 0, stream>>>(
        x, out, (int)tiles);
  }
  const long long tailStart = tiles * SPX_TROWS;
  if (tailStart < rows) {
    sparsemax_rowwave_kernel<<<dim3(64), dim3(SPX_TPB), 0, stream>>>(
        x, out, tailStart, rows);
  }
}